// SelfAttention_12575664242746
// MI455X (gfx1250) — compile-verified
//
#include <hip/hip_runtime.h>
#include <math.h>

typedef float v2f __attribute__((ext_vector_type(2)));
typedef float v8f __attribute__((ext_vector_type(8)));

#define N_TOK 8192
#define DDIM  256
#define TJ    32   // K/V rows per streamed tile

__device__ __forceinline__ v8f wmma_f32_16x16x4(v2f a, v2f b, v8f c) {
  return __builtin_amdgcn_wmma_f32_16x16x4_f32(false, a, false, b, (short)0, c,
                                               false, false);
}

__device__ __forceinline__ v8f vzero8() {
  v8f z = {0.f, 0.f, 0.f, 0.f, 0.f, 0.f, 0.f, 0.f};
  return z;
}

// ---------------------------------------------------------------------------
// Kernel 1: fused QKV projection.  out = x @ W + b   for W in {Wq, Wk, Wv}
// grid = (N/16 row tiles, 3 matrices), block = 128 (4 waves)
// ---------------------------------------------------------------------------
__global__ __launch_bounds__(128) void qkv_proj_kernel(
    const float* __restrict__ x,
    const float* __restrict__ Wq, const float* __restrict__ bq,
    const float* __restrict__ Wk, const float* __restrict__ bk,
    const float* __restrict__ Wv, const float* __restrict__ bv,
    float* __restrict__ qo, float* __restrict__ ko, float* __restrict__ vo) {
  const float* W;
  const float* bias;
  float* out;
  if (blockIdx.y == 0) {
    W = Wq; bias = bq; out = qo;
  } else if (blockIdx.y == 1) {
    W = Wk; bias = bk; out = ko;
  } else {
    W = Wv; bias = bv; out = vo;
  }

  const int row0 = blockIdx.x * 16;
  const int wave = threadIdx.x >> 5;
  const int lane = threadIdx.x & 31;
  const int ln   = lane & 15;
  const int kh   = (lane >> 4) << 1;  // 0 or 2
  const int mb   = (lane >> 4) << 3;  // 0 or 8
  const int col0 = wave * 64;

  v8f acc[4];
#pragma unroll
  for (int t = 0; t < 4; ++t) acc[t] = vzero8();

  const float* xrow = x + (size_t)(row0 + ln) * DDIM + kh;
#pragma unroll 4
  for (int c = 0; c < DDIM; c += 4) {
    v2f a = *(const v2f*)(xrow + c);
#pragma unroll
    for (int t = 0; t < 4; ++t) {
      const float* wp = W + (size_t)(c + kh) * DDIM + col0 + t * 16 + ln;
      v2f bfr;
      bfr.x = wp[0];
      bfr.y = wp[DDIM];
      acc[t] = wmma_f32_16x16x4(a, bfr, acc[t]);
    }
  }

#pragma unroll
  for (int t = 0; t < 4; ++t) {
    const int col = col0 + t * 16 + ln;
    const float bb = bias[col];
#pragma unroll
    for (int r = 0; r < 8; ++r)
      out[(size_t)(row0 + mb + r) * DDIM + col] = acc[t][r] + bb;
  }
}

// ---------------------------------------------------------------------------
// Kernel 2: flash attention, fp32 WMMA, double-buffered async K/V staging.
// grid = N/64, block = 128 (4 waves). Wave w owns 16 Q rows.
// ---------------------------------------------------------------------------

// LDS layout (floats):
//   K buffers: [0 .. 8191] and [8192 .. 16383]       (TJ x 256 each)
//   V buffers: [16384 .. 24575] and [24576 .. 32767]
//   P scratch: [32768 .. 34815]  (4 waves x 16 x 32)
#define LDS_KOFF(b) ((b) * (TJ * DDIM))
#define LDS_VOFF(b) (2 * (TJ * DDIM) + (b) * (TJ * DDIM))
#define LDS_POFF    (4 * (TJ * DDIM))
#define LDS_FLOATS  (4 * (TJ * DDIM) + 4 * 16 * TJ)

// Issue an async global->LDS copy of one TJ x 256 fp32 tile (32 KB).
// 128 threads x 16 b128 transfers each, ASYNCcnt-tracked.
__device__ __forceinline__ void async_tile(const float* gsrc,
                                           unsigned lds_byte_addr, int tid) {
  unsigned long long g =
      (unsigned long long)(uintptr_t)gsrc + (unsigned long long)tid * 16ull;
  unsigned l = lds_byte_addr + (unsigned)tid * 16u;
#pragma unroll
  for (int i = 0; i < (TJ * DDIM * 4) / (128 * 16); ++i) {
    asm volatile("global_load_async_to_lds_b128 %0, %1, off"
                 :
                 : "v"(l), "v"(g)
                 : "memory");
    l += 128u * 16u;
    g += 128ull * 16ull;
  }
}

__global__ __launch_bounds__(128, 1) void flash_attn_kernel(
    const float* __restrict__ q, const float* __restrict__ k,
    const float* __restrict__ v, float* __restrict__ out) {
  __shared__ float smem[LDS_FLOATS];  // 136 KB

  const int tid  = threadIdx.x;
  const int wave = tid >> 5;
  const int lane = tid & 31;
  const int ln   = lane & 15;
  const int kh   = (lane >> 4) << 1;  // 0 or 2
  const int mb   = (lane >> 4) << 3;  // 0 or 8
  const int row0 = blockIdx.x * 64 + wave * 16;

  const unsigned lds_base = (unsigned)(uintptr_t)&smem[0];

  // Preload Q tile as A-fragments, pre-scaled by 1/sqrt(D).
  v2f Qa[64];
  const float* qrow = q + (size_t)(row0 + ln) * DDIM + kh;
#pragma unroll
  for (int c = 0; c < 64; ++c) {
    v2f a = *(const v2f*)(qrow + 4 * c);
    Qa[c] = a * 0.0625f;
  }

  v8f Oacc[16];
#pragma unroll
  for (int t = 0; t < 16; ++t) Oacc[t] = vzero8();

  float mrun[8], lrun[8];
#pragma unroll
  for (int r = 0; r < 8; ++r) {
    mrun[r] = -3.0e38f;
    lrun[r] = 0.f;
  }

  float* pbuf = smem + LDS_POFF + wave * (16 * TJ);

  // Prologue: stage tile 0 into buffer 0.
  async_tile(k, lds_base + LDS_KOFF(0) * 4, tid);
  async_tile(v, lds_base + LDS_VOFF(0) * 4, tid);

  const int NIT = N_TOK / TJ;
  for (int it = 0; it < NIT; ++it) {
    // Wait for this wave's async fills of the current tile, then sync all.
    asm volatile("s_wait_asynccnt 0x0" ::: "memory");
    __syncthreads();

    // Kick off next tile into the other buffer (overlaps with compute).
    if (it + 1 < NIT) {
      const int nb = (it + 1) & 1;
      const size_t joff = (size_t)(it + 1) * TJ * DDIM;
      async_tile(k + joff, lds_base + LDS_KOFF(nb) * 4, tid);
      async_tile(v + joff, lds_base + LDS_VOFF(nb) * 4, tid);
    }

    const int cb = it & 1;
    const float* Kbuf = smem + LDS_KOFF(cb);
    const float* Vbuf = smem + LDS_VOFF(cb);

    // ---- S = Qs @ K_tile^T : two 16x16 tiles, 128 WMMAs ----
    v8f s0 = vzero8(), s1 = vzero8();
#pragma unroll
    for (int c = 0; c < 64; ++c) {
      v2f b0 = *(const v2f*)(&Kbuf[ln * DDIM + 4 * c + kh]);
      v2f b1 = *(const v2f*)(&Kbuf[(16 + ln) * DDIM + 4 * c + kh]);
      s0 = wmma_f32_16x16x4(Qa[c], b0, s0);
      s1 = wmma_f32_16x16x4(Qa[c], b1, s1);
    }

    // ---- online softmax over 32 columns ----
    float rowmax[8];
#pragma unroll
    for (int r = 0; r < 8; ++r) {
      float t0 = fmaxf(s0[r], s1[r]);
      t0 = fmaxf(t0, __shfl_xor(t0, 1));
      t0 = fmaxf(t0, __shfl_xor(t0, 2));
      t0 = fmaxf(t0, __shfl_xor(t0, 4));
      t0 = fmaxf(t0, __shfl_xor(t0, 8));
      rowmax[r] = t0;
    }

    float alpha[8];
#pragma unroll
    for (int r = 0; r < 8; ++r) {
      const float mnew = fmaxf(mrun[r], rowmax[r]);
      alpha[r] = __expf(mrun[r] - mnew);
      mrun[r] = mnew;
    }

#pragma unroll
    for (int r = 0; r < 8; ++r) {
      const float p0 = __expf(s0[r] - mrun[r]);
      const float p1 = __expf(s1[r] - mrun[r]);
      s0[r] = p0;
      s1[r] = p1;
      float t0 = p0 + p1;
      t0 += __shfl_xor(t0, 1);
      t0 += __shfl_xor(t0, 2);
      t0 += __shfl_xor(t0, 4);
      t0 += __shfl_xor(t0, 8);
      lrun[r] = lrun[r] * alpha[r] + t0;
    }

#pragma unroll
    for (int t = 0; t < 16; ++t)
#pragma unroll
      for (int r = 0; r < 8; ++r) Oacc[t][r] *= alpha[r];

    // ---- C-layout P -> A-layout via per-wave LDS scratch (16 x 32) ----
#pragma unroll
    for (int r = 0; r < 8; ++r) {
      pbuf[(mb + r) * TJ + ln] = s0[r];
      pbuf[(mb + r) * TJ + 16 + ln] = s1[r];
    }
    asm volatile("s_wait_dscnt 0x0" ::: "memory");

    // ---- O += P @ V_tile : 8 chunks x 16 col tiles = 128 WMMAs ----
#pragma unroll
    for (int c = 0; c < TJ / 4; ++c) {
      v2f pa = *(const v2f*)(&pbuf[ln * TJ + 4 * c + kh]);
      const float* vp0 = &Vbuf[(4 * c + kh) * DDIM + ln];
#pragma unroll
      for (int t = 0; t < 16; ++t) {
        v2f bfr;
        bfr.x = vp0[t * 16];
        bfr.y = vp0[t * 16 + DDIM];
        Oacc[t] = wmma_f32_16x16x4(pa, bfr, Oacc[t]);
      }
    }

    // All waves done with current buffers before tile it+2 overwrites them.
    __syncthreads();
  }

  // ---- epilogue: O / l ----
#pragma unroll
  for (int t = 0; t < 16; ++t) {
    const int col = t * 16 + ln;
#pragma unroll
    for (int r = 0; r < 8; ++r) {
      out[(size_t)(row0 + mb + r) * DDIM + col] = Oacc[t][r] / lrun[r];
    }
  }
}

// ---------------------------------------------------------------------------
extern "C" void kernel_launch(void* const* d_in, const int* in_sizes, int n_in,
                              void* d_out, int out_size, void* d_ws,
                              size_t ws_size, hipStream_t stream) {
  const float* x  = (const float*)d_in[0];
  const float* Wq = (const float*)d_in[1];
  const float* bq = (const float*)d_in[2];
  const float* Wk = (const float*)d_in[3];
  const float* bk = (const float*)d_in[4];
  const float* Wv = (const float*)d_in[5];
  const float* bv = (const float*)d_in[6];
  float* out = (float*)d_out;

  float* qbuf = (float*)d_ws;                 // 8 MB
  float* kbuf = qbuf + (size_t)N_TOK * DDIM;  // 8 MB
  float* vbuf = kbuf + (size_t)N_TOK * DDIM;  // 8 MB

  dim3 g1(N_TOK / 16, 3);
  qkv_proj_kernel<<<g1, 128, 0, stream>>>(x, Wq, bq, Wk, bk, Wv, bv, qbuf,
                                          kbuf, vbuf);
  flash_attn_kernel<<<N_TOK / 64, 128, 0, stream>>>(qbuf, kbuf, vbuf, out);
}